// GridEncoder_10754598109278
// MI455X (gfx1250) — compile-verified
//
#include <hip/hip_runtime.h>
#include <stdint.h>

// ---------------------------------------------------------------------------
// GridEncoder (instant-NGP hash grid) forward for MI455X / gfx1250.
// D=3, L=16, C=2, H=16, per_level_scale=2, log2_hashmap=19, align_corners=0.
//
// Level meta (precomputed to match the reference's _level_meta()):
//   l=0: res=16  dense size=4920    off=0
//   l=1: res=32  dense size=35944   off=4920
//   l=2: res=64  dense size=274632  off=40864
//   l>=3: hashed size=524288        off=315496 + (l-3)*524288
// Total params = 7,131,240 entries (float2) ~= 57 MB  -> L2 resident.
// ---------------------------------------------------------------------------

typedef uint32_t u32x4 __attribute__((ext_vector_type(4)));
typedef int32_t  i32x8 __attribute__((ext_vector_type(8)));
typedef int32_t  i32x4 __attribute__((ext_vector_type(4)));
typedef float    f32x4 __attribute__((ext_vector_type(4)));

static __device__ __constant__ const uint32_t kOff[16] = {
    0u,       4920u,    40864u,   315496u,  839784u,  1364072u,
    1888360u, 2412648u, 2936936u, 3461224u, 3985512u, 4509800u,
    5034088u, 5558376u, 6082664u, 6606952u};

#define PRIME_Y 2654435761u
#define PRIME_Z 805459861u
#define HASH_MASK 0x7FFFFu
#define L0_ENTRIES 4920

__global__ __launch_bounds__(256) void grid_encode_fwd(
    const float* __restrict__ xin, const float2* __restrict__ emb,
    float* __restrict__ out, int n) {
  // Level-0 dense table lives in LDS for the whole block (39,360 bytes).
  __shared__ float2 s_lut[L0_ENTRIES];

  // ---- TDM: DMA the level-0 table global -> LDS (wave 0 only issues; TDM
  // ignores EXEC, so guard with a whole-wave branch to avoid duplicates). ----
  if (threadIdx.x < 32) {
    uint64_t ga = (uint64_t)(uintptr_t)emb;           // level-0 offset == 0
    uint32_t lds_addr = (uint32_t)(uintptr_t)&s_lut[0];

    u32x4 g0;
    g0[0] = 1u;                                       // count=1 (valid), user mode
    g0[1] = lds_addr;                                 // lds_addr[31:0]
    g0[2] = (uint32_t)(ga & 0xFFFFFFFFu);             // global_addr[31:0]
    g0[3] = (uint32_t)((ga >> 32) & 0x1FFFFFFu)       // global_addr[56:32]
          | (2u << 30);                               // type = 2 ("image")

    i32x8 g1;
    g1[0] = (int)(3u << 16);                          // data_size=3 (8B), mask=0
    g1[1] = (int)((uint32_t)L0_ENTRIES << 16);        // tensor_dim0[15:0]
    g1[2] = (int)(1u << 16);                          // tensor_dim0 hi=0, dim1=1
    g1[3] = (int)((uint32_t)L0_ENTRIES << 16);        // dim1 hi=0, tile_dim0=4920
    g1[4] = 1;                                        // tile_dim1=1, tile_dim2=0
    g1[5] = (int)L0_ENTRIES;                          // tensor_dim0_stride lo
    g1[6] = 0;                                        // stride hi / dim1_stride lo
    g1[7] = 0;                                        // dim1_stride hi
    i32x4 gz4 = {0, 0, 0, 0};                         // groups 2/3 unused (2D)
    i32x8 gz8 = {0, 0, 0, 0, 0, 0, 0, 0};             // trailing group (clang-23 form)

    __builtin_amdgcn_tensor_load_to_lds(g0, g1, gz4, gz4, gz8, 0);
    __builtin_amdgcn_s_wait_tensorcnt(0);             // s_wait_tensorcnt 0
  }
  __syncthreads();

  int b = blockIdx.x * blockDim.x + threadIdx.x;
  if (b >= n) return;

  // Streamed input: non-temporal loads, do not pollute L2 (table lives there).
  const float* xp = xin + (size_t)b * 3;
  float x0 = (__builtin_nontemporal_load(xp + 0) + 1.0f) * 0.5f;
  float y0 = (__builtin_nontemporal_load(xp + 1) + 1.0f) * 0.5f;
  float z0 = (__builtin_nontemporal_load(xp + 2) + 1.0f) * 0.5f;

  f32x4 o[8];

#pragma unroll
  for (int l = 0; l < 16; ++l) {
    const uint32_t res = 16u << l;
    const float scale = (float)(res - 1u);            // exp2(l)*H - 1
    float px = fmaf(x0, scale, 0.5f);
    float py = fmaf(y0, scale, 0.5f);
    float pz = fmaf(z0, scale, 0.5f);
    float flx = floorf(px), fly = floorf(py), flz = floorf(pz);
    uint32_t gx = (uint32_t)flx, gy = (uint32_t)fly, gz = (uint32_t)flz;
    float fx = px - flx, fy = py - fly, fz = pz - flz;
    float ux = 1.0f - fx, uy = 1.0f - fy, uz = 1.0f - fz;

    // Corner index c: bit0 -> x, bit1 -> y, bit2 -> z (matches reference).
    uint32_t i0, i1, i2, i3, i4, i5, i6, i7;
    if (l >= 3) {
      // fast_hash: x*1 ^ y*P1 ^ z*P2, masked to 2^19 entries.
      uint32_t ax0 = gx, ax1 = gx + 1u;
      uint32_t ay0 = gy * PRIME_Y, ay1 = ay0 + PRIME_Y;
      uint32_t az0 = gz * PRIME_Z, az1 = az0 + PRIME_Z;
      i0 = (ax0 ^ ay0 ^ az0) & HASH_MASK;
      i1 = (ax1 ^ ay0 ^ az0) & HASH_MASK;
      i2 = (ax0 ^ ay1 ^ az0) & HASH_MASK;
      i3 = (ax1 ^ ay1 ^ az0) & HASH_MASK;
      i4 = (ax0 ^ ay0 ^ az1) & HASH_MASK;
      i5 = (ax1 ^ ay0 ^ az1) & HASH_MASK;
      i6 = (ax0 ^ ay1 ^ az1) & HASH_MASK;
      i7 = (ax1 ^ ay1 ^ az1) & HASH_MASK;
    } else {
      // Dense: idx = x + y*(res+1) + z*(res+1)^2; always < padded size,
      // so the reference's % size is a no-op here.
      uint32_t s1 = res + 1u, s2 = s1 * s1;
      uint32_t c = gx + gy * s1 + gz * s2;
      i0 = c;            i1 = c + 1u;
      i2 = c + s1;       i3 = c + s1 + 1u;
      i4 = c + s2;       i5 = c + s2 + 1u;
      i6 = c + s2 + s1;  i7 = c + s2 + s1 + 1u;
    }

    // Issue all 8 gathers, then consume (compiler batches loadcnt waits).
    float2 e0, e1, e2, e3, e4, e5, e6, e7;
    if (l == 0) {
      e0 = s_lut[i0]; e1 = s_lut[i1]; e2 = s_lut[i2]; e3 = s_lut[i3];
      e4 = s_lut[i4]; e5 = s_lut[i5]; e6 = s_lut[i6]; e7 = s_lut[i7];
    } else {
      const float2* __restrict__ t = emb + kOff[l];
      e0 = t[i0]; e1 = t[i1]; e2 = t[i2]; e3 = t[i3];
      e4 = t[i4]; e5 = t[i5]; e6 = t[i6]; e7 = t[i7];
    }

    float w0 = ux * uy * uz, w1 = fx * uy * uz;
    float w2 = ux * fy * uz, w3 = fx * fy * uz;
    float w4 = ux * uy * fz, w5 = fx * uy * fz;
    float w6 = ux * fy * fz, w7 = fx * fy * fz;

    float ox = fmaf(w0, e0.x, fmaf(w1, e1.x, fmaf(w2, e2.x, fmaf(w3, e3.x,
               fmaf(w4, e4.x, fmaf(w5, e5.x, fmaf(w6, e6.x, w7 * e7.x)))))));
    float oy = fmaf(w0, e0.y, fmaf(w1, e1.y, fmaf(w2, e2.y, fmaf(w3, e3.y,
               fmaf(w4, e4.y, fmaf(w5, e5.y, fmaf(w6, e6.y, w7 * e7.y)))))));

    if (l & 1) { o[l >> 1].z = ox; o[l >> 1].w = oy; }
    else       { o[l >> 1].x = ox; o[l >> 1].y = oy; }
  }

  // Streamed output: 8x non-temporal b128 stores (128 B/thread, 16B aligned).
  f32x4* ov = (f32x4*)(out + (size_t)b * 32);
#pragma unroll
  for (int i = 0; i < 8; ++i) __builtin_nontemporal_store(o[i], ov + i);
}

extern "C" void kernel_launch(void* const* d_in, const int* in_sizes, int n_in,
                              void* d_out, int out_size, void* d_ws,
                              size_t ws_size, hipStream_t stream) {
  (void)n_in; (void)out_size; (void)d_ws; (void)ws_size;
  const float*  x   = (const float*)d_in[0];
  const float2* emb = (const float2*)d_in[1];
  float*        out = (float*)d_out;
  int n = in_sizes[0] / 3;                 // 1,048,576 points
  const int block = 256;                   // 8 wave32s per block
  int grid = (n + block - 1) / block;
  grid_encode_fwd<<<grid, block, 0, stream>>>(x, emb, out, n);
}